// DbrxRouter_49228915147013
// MI455X (gfx1250) — compile-verified
//
#include <hip/hip_runtime.h>

// DBRX MoE router for MI455X (gfx1250, wave32).
// logits = x[N,6144] @ W[16,6144]^T via V_WMMA_F32_16X16X4_F32,
// then fp32 softmax(16), top-4, L1-normalize.
// Memory-bound: 403MB of x reads -> ~17us floor @ 23.3TB/s, requiring
// ~186 TFLOP/s fp32 -> use the fp32 matrix pipe, not plain VALU.

#define HIDDEN 6144
#define NEXP 16
#define TOPK 4
#define KT 128                     // K tile per LDS stage
#define LDS_STRIDE (KT + 4)        // 132 floats = 528B row pitch (16B aligned, bank-staggered)
#define TOK_PER_BLOCK 128
#define THREADS 256                // 8 waves (wave32)

typedef __attribute__((ext_vector_type(2))) float v2f;
typedef __attribute__((ext_vector_type(8))) float v8f;

__global__ __launch_bounds__(THREADS) void dbrx_router_kernel(
    const float* __restrict__ x, const float* __restrict__ W,
    float* __restrict__ out, int Ntok)
{
    __shared__ float ldsA[TOK_PER_BLOCK * LDS_STRIDE]; // 128*132*4 = 67584 B
    __shared__ float ldsW[NEXP * LDS_STRIDE];          // 16*132*4  =  8448 B

    const int tid  = threadIdx.x;
    const int wave = tid >> 5;        // 0..7
    const int lane = tid & 31;
    const int n    = lane & 15;       // expert col (B) / token row (A) within tile
    const int h    = lane >> 4;       // half-wave: selects K pair {2h, 2h+1}
    const int tok0 = blockIdx.x * TOK_PER_BLOCK;

    v8f c = {}; // 16x16 f32 accumulator (logits tile: 16 tokens x 16 experts)

    for (int k0 = 0; k0 < HIDDEN; k0 += KT) {
        // ---- stage A tile: 128 tokens x 128 K floats, coalesced b128 loads ----
        // one wave covers one contiguous 512B row segment per pass
        #pragma unroll
        for (int p = 0; p < 16; ++p) {
            int f   = p * THREADS + tid;   // float4 slot id in tile (4096 total)
            int row = f >> 5;              // token row 0..127
            int c4  = f & 31;              // float4 col 0..31
            float4 v = *(const float4*)(x + (size_t)(tok0 + row) * HIDDEN + k0 + c4 * 4);
            *(float4*)(&ldsA[row * LDS_STRIDE + c4 * 4]) = v;
        }
        // ---- stage W tile: 16 experts x 128 K floats ----
        #pragma unroll
        for (int p = 0; p < 2; ++p) {
            int f   = p * THREADS + tid;
            int row = f >> 5;              // expert 0..15
            int c4  = f & 31;
            float4 v = *(const float4*)(W + (size_t)row * HIDDEN + k0 + c4 * 4);
            *(float4*)(&ldsW[row * LDS_STRIDE + c4 * 4]) = v;
        }
        // prefetch next A tile into cache hierarchy (global_prefetch_b8)
        if (k0 + KT < HIDDEN) {
            int row = tid >> 5, c4 = tid & 31;
            __builtin_prefetch(x + (size_t)(tok0 + row) * HIDDEN + (k0 + KT) + c4 * 4, 0, 0);
        }
        __syncthreads();

        // ---- 32 x V_WMMA_F32_16X16X4_F32 over this K tile ----
        // A frag: lane holds row M=n, K={k+2h, k+2h+1}  -> ds_load_b64
        // B frag: lane holds col N=n (expert), same K pair -> ds_load_b64
        const float* aBase = &ldsA[(wave * 16 + n) * LDS_STRIDE + 2 * h];
        const float* bBase = &ldsW[n * LDS_STRIDE + 2 * h];
        #pragma unroll
        for (int s = 0; s < KT / 4; ++s) {
            v2f a = *(const v2f*)(aBase + s * 4);
            v2f b = *(const v2f*)(bBase + s * 4);
            c = __builtin_amdgcn_wmma_f32_16x16x4_f32(
                    false, a, false, b, (short)0, c, false, false);
        }
        __syncthreads();
    }

    // ---- scatter logits tile to LDS [tokenLocal][expert] (reuse ldsA) ----
    // C layout: VGPR r -> M = r (lanes 0-15) / r+8 (lanes 16-31), N = n
    float* logits = ldsA;
    #pragma unroll
    for (int r = 0; r < 8; ++r) {
        int tokLocal = wave * 16 + h * 8 + r;
        logits[tokLocal * NEXP + n] = c[r];
    }
    __syncthreads();

    // ---- per-token epilogue: softmax(16) + top-4 + L1 normalize ----
    if (tid < TOK_PER_BLOCK) {
        int tok = tok0 + tid;
        float l[NEXP];
        float m = -1e30f;
        #pragma unroll
        for (int e = 0; e < NEXP; ++e) {
            l[e] = logits[tid * NEXP + e];
            m = fmaxf(m, l[e]);
        }
        float sum = 0.f;
        #pragma unroll
        for (int e = 0; e < NEXP; ++e) {
            l[e] = __expf(l[e] - m);   // v_exp_f32
            sum += l[e];
        }
        float inv = 1.0f / sum;
        float w[NEXP];
        #pragma unroll
        for (int e = 0; e < NEXP; ++e) w[e] = l[e] * inv;

        // output 0: full softmax weights [N,16]
        float4* wp = (float4*)(out + (size_t)tok * NEXP);
        #pragma unroll
        for (int q = 0; q < 4; ++q) {
            float4 v = { w[4*q+0], w[4*q+1], w[4*q+2], w[4*q+3] };
            wp[q] = v;
        }

        // top-4: 4 argmax passes, strict '>' => lowest index wins ties (jax-stable)
        float tmp[NEXP];
        #pragma unroll
        for (int e = 0; e < NEXP; ++e) tmp[e] = w[e];
        float tw[TOPK]; int te[TOPK]; float tsum = 0.f;
        #pragma unroll
        for (int j = 0; j < TOPK; ++j) {
            float best = -1.0f; int bi = 0;
            #pragma unroll
            for (int e = 0; e < NEXP; ++e)
                if (tmp[e] > best) { best = tmp[e]; bi = e; }
            tw[j] = best; te[j] = bi; tmp[bi] = -1.0f; tsum += best;
        }
        float tinv = 1.0f / tsum;  // L1 norm of positive softmax weights = sum

        float* outTW = out + (size_t)Ntok * NEXP;
        float* outTE = out + (size_t)Ntok * (NEXP + TOPK);
        float4 twv = { tw[0]*tinv, tw[1]*tinv, tw[2]*tinv, tw[3]*tinv };
        *(float4*)(outTW + (size_t)tok * TOPK) = twv;
        // expert indices written as float values into the fp32 output buffer
        float4 tev = { (float)te[0], (float)te[1], (float)te[2], (float)te[3] };
        *(float4*)(outTE + (size_t)tok * TOPK) = tev;
    }
}

extern "C" void kernel_launch(void* const* d_in, const int* in_sizes, int n_in,
                              void* d_out, int out_size, void* d_ws, size_t ws_size,
                              hipStream_t stream) {
    const float* x = (const float*)d_in[0];   // [4,4096,6144] fp32
    const float* W = (const float*)d_in[1];   // [16,6144] fp32
    float* out = (float*)d_out;               // weights | top_weights | top_experts
    int Ntok = in_sizes[0] / HIDDEN;          // 16384
    int blocks = Ntok / TOK_PER_BLOCK;        // 128
    dbrx_router_kernel<<<blocks, THREADS, 0, stream>>>(x, W, out, Ntok);
}